// AdaptiveTokenMixer_26663156973946
// MI455X (gfx1250) — compile-verified
//
#include <hip/hip_runtime.h>
#include <hip/hip_bf16.h>

typedef __attribute__((ext_vector_type(2))) float v2f;
typedef __attribute__((ext_vector_type(8))) float v8f;

#define NEGV (-1e9f)
#define DDIM 256                  // feature dim (hardcoded -> immediate offsets)
#define ROWB (DDIM * 4)           // 1024 bytes per token row

__device__ __forceinline__ void load_bfrags(v2f* bf, const char* xb,
                                            const unsigned* rowoff, int dc) {
#pragma unroll
    for (int jb = 0; jb < 6; ++jb) {
        // B 4x16 f32: lanes 0-15 hold K={0,1} N=lane; lanes 16-31 K={2,3} N=lane-16.
        bf[jb].x = *(const float*)(xb + rowoff[2 * jb + 0] + dc * 64);
        bf[jb].y = *(const float*)(xb + rowoff[2 * jb + 1] + dc * 64);
    }
}

__global__ __launch_bounds__(256) void atm_wmma_kernel(
    const float* __restrict__ x,          // [B,N,256]
    const float* __restrict__ dts,        // [B,N]
    const unsigned char* __restrict__ vm, // [B,N] bool (1 byte per elem)
    const float* __restrict__ w,          // [8]
    const float* __restrict__ beta,       // [1]
    float* __restrict__ out,              // [B,N,256]
    int N)
{
    __shared__ float alphaLds[128 * 8];

    const int tid = threadIdx.x;
    const int tilesPerB = N >> 7;                 // N / 128
    const int bidx = blockIdx.x / tilesPerB;
    const int n0   = (blockIdx.x % tilesPerB) << 7;

    // softmax(w), sigmoid(beta): tiny, recomputed per thread
    float wraw[8];
#pragma unroll
    for (int k = 0; k < 8; ++k) wraw[k] = w[k];
    float wmax = wraw[0];
#pragma unroll
    for (int k = 1; k < 8; ++k) wmax = fmaxf(wmax, wraw[k]);
    float wsm[8], wsum = 0.f;
#pragma unroll
    for (int k = 0; k < 8; ++k) { wsm[k] = __expf(wraw[k] - wmax); wsum += wsm[k]; }
    const float winv = 1.f / wsum;
#pragma unroll
    for (int k = 0; k < 8; ++k) wsm[k] *= winv;
    const float bsig = 1.f / (1.f + __expf(-beta[0]));

    // --- Phase 1: per-token alpha[8] -> LDS ---
    if (tid < 128) {
        const int n = n0 + tid;
        const long tb = (long)bidx * N;
        const float dt0 = dts[tb + n];
        const bool v0 = (vm[tb + n] != 0);
        float logit[8];
        bool  vk[8];
#pragma unroll
        for (int k = 0; k < 8; ++k) {
            const int idx = n + k;
            bool val; float td;
            if (k == 0)       { val = v0; td = 0.f; }
            else if (idx < N) { val = v0 && (vm[tb + idx] != 0);
                                td = fmaxf(dts[tb + idx] - dt0, 0.f); }
            else              { val = false; td = 0.f; }
            vk[k] = val;
            logit[k] = val ? -td : NEGV;
        }
        float m = logit[0];
#pragma unroll
        for (int k = 1; k < 8; ++k) m = fmaxf(m, logit[k]);
        float es[8], s = 0.f;
#pragma unroll
        for (int k = 0; k < 8; ++k) { es[k] = __expf(logit[k] - m); s += es[k]; }
        const float sinv = 1.f / s;
        float a[8], asum = 0.f;
#pragma unroll
        for (int k = 0; k < 8; ++k) {
            const float theta = es[k] * sinv;
            const float av = vk[k] ? (bsig * wsm[k] + (1.f - bsig) * theta) : 0.f;
            a[k] = av; asum += av;
        }
        const float ainv = 1.f / fmaxf(asum, 1e-8f);
#pragma unroll
        for (int k = 0; k < 8; ++k) alphaLds[tid * 8 + k] = a[k] * ainv;
    }
    __syncthreads();

    // --- Phase 2: banded matmul via WMMA f32 16x16x4, one 16-token subtile per wave ---
    const int lane = tid & 31;
    const int wave = tid >> 5;
    const int m0l  = wave << 4;
    const int i    = lane & 15;
    const int hi   = lane >> 4;

    // A 16x4 f32: lanes 0-15 hold M=lane K={0,1}; lanes 16-31 hold M=lane-16 K={2,3}.
    v2f afrag[6];
#pragma unroll
    for (int jb = 0; jb < 6; ++jb) {
        const int jbase = 4 * jb + 2 * hi;
        const int k0 = jbase - i;
        const int k1 = k0 + 1;
        afrag[jb].x = (k0 >= 0 && k0 < 8) ? alphaLds[(m0l + i) * 8 + k0] : 0.f;
        afrag[jb].y = (k1 >= 0 && k1 < 8) ? alphaLds[(m0l + i) * 8 + k1] : 0.f;
    }

    // Uniform per-batch slab base (SGPR); all lane offsets fit u32 (4 MB slab).
    const char* xb = (const char*)(x   + (size_t)bidx * N * DDIM);
    char*       ob = (char*)      (out + (size_t)bidx * N * DDIM);
    const int gr0 = n0 + m0l;

    // Precompute the 12 clamped B-row byte offsets once (clamp handles the grid
    // tail uniformly: clamped rows always carry zero alpha coefficients).
    unsigned rowoff[12];
#pragma unroll
    for (int jb = 0; jb < 6; ++jb) {
#pragma unroll
        for (int sel = 0; sel < 2; ++sel) {
            unsigned r = (unsigned)(gr0 + 4 * jb + 2 * hi + sel);
            r = (r < (unsigned)N) ? r : (unsigned)(N - 1);
            rowoff[2 * jb + sel] = r * ROWB + (unsigned)i * 4u;
        }
    }
    // Store base: D 16x16 -> row gr0 + j + 8*hi, col = dc*16 + i (immediates j*1024 + dc*64).
    const unsigned soff = (unsigned)(gr0 + 8 * hi) * ROWB + (unsigned)i * 4u;

    v2f cur[6], nxt[6];
    load_bfrags(cur, xb, rowoff, 0);

#pragma unroll
    for (int dc = 0; dc < 16; ++dc) {
        if (dc < 15) load_bfrags(nxt, xb, rowoff, dc + 1);   // prefetch next chunk
        v8f acc = {};
#pragma unroll
        for (int jb = 0; jb < 6; ++jb) {
            acc = __builtin_amdgcn_wmma_f32_16x16x4_f32(
                false, afrag[jb], false, cur[jb], (short)0, acc, false, false);
        }
#pragma unroll
        for (int j = 0; j < 8; ++j) {
            *(float*)(ob + soff + j * ROWB + dc * 64) = acc[j];
        }
#pragma unroll
        for (int jb = 0; jb < 6; ++jb) cur[jb] = nxt[jb];
    }
}

extern "C" void kernel_launch(void* const* d_in, const int* in_sizes, int n_in,
                              void* d_out, int out_size, void* d_ws, size_t ws_size,
                              hipStream_t stream) {
    const float*         x    = (const float*)d_in[0];
    const float*         dts  = (const float*)d_in[1];
    const unsigned char* vm   = (const unsigned char*)d_in[2]; // JAX bool -> 1 byte
    const float*         w    = (const float*)d_in[3];
    const float*         beta = (const float*)d_in[4];
    float*               out  = (float*)d_out;

    const int BN = in_sizes[1];          // B*N
    const int B  = 8;
    const int N  = BN / B;               // 4096

    const int blocks = B * (N >> 7);     // 128 tokens per block
    atm_wmma_kernel<<<blocks, 256, 0, stream>>>(x, dts, vm, w, beta, out, N);
}